// O3TensorProduct_87909390614541
// MI455X (gfx1250) — compile-verified
//
#include <hip/hip_runtime.h>

typedef float v2f __attribute__((ext_vector_type(2)));
typedef float v8f __attribute__((ext_vector_type(8)));

#define WAVES 4
#define BLOCK (WAVES * 32)
#define ROWS_PER_BLOCK (WAVES * 16)

// LDS plan (floats) -- all weight matrices stored TRANSPOSED with padded K-stride so a
// B-fragment (rows k,k+1 of one column) is a single bank-conflict-free ds_load_b64:
//   wgtACT : 96 cols x 68  = 6528   ([C_SS0*W_ss0 | kSV1*W_sv1]^T, [w][u])
//   wgtBT  : 64 cols x 36  = 2304   ((kVV0*W_vv0)^T)
//   wgtDT  : 32 cols x 36  = 1152   ((kVS1*W_vs1)^T)
// per wave: actS1 16x68 (1088) | actD 16x36 (576) | actV 3x16x36 (1728) | v2t 16x4 (64)
// total = 9984 + WAVES*3456 floats = 95,232 B (gfx1250 LDS cap > 64 KB confirmed)

__global__ __launch_bounds__(BLOCK) void o3tp_wmma_f32_kernel(
    const float* __restrict__ x1, const float* __restrict__ x2,
    const float* __restrict__ Wss0, const float* __restrict__ Wvv0,
    const float* __restrict__ Wsv1, const float* __restrict__ Wvs1,
    const float* __restrict__ bias0, float* __restrict__ out, int nrows)
{
    __shared__ float lds[9984 + WAVES * 3456];
    float* wgtACT = lds;          // 6528
    float* wgtBT  = lds + 6528;   // 2304
    float* wgtDT  = lds + 8832;   // 1152

    const int tid  = threadIdx.x;
    const int wave = tid >> 5;
    const int lane = tid & 31;
    const int h    = lane >> 4;   // lane half (0/1)
    const int lr   = lane & 15;   // row within tile / column within N-tile

    float* wbase = lds + 9984 + wave * 3456;
    float* actS1 = wbase;          // 16 x 68
    float* actD  = wbase + 1088;   // 16 x 36
    float* actV  = wbase + 1664;   // 3 x 16 x 36
    float* v2t   = wbase + 3392;   // 16 x 4  (s2, v2x, v2y, v2z per row)

    constexpr float kSS0 = 0.08838834764831845f;  // sqrt(1/128)
    constexpr float kVV0 = 0.07216878364870323f;  // sqrt(1/64)/sqrt(3)
    constexpr float kSV1 = 0.08838834764831845f;  // sqrt(3/128)/sqrt(3) = sqrt(1/128)
    constexpr float kVS1 = 0.125f;                // sqrt(3/64)/sqrt(3)  = sqrt(1/64)

    // ---- Phase 0: cooperative weight staging (transpose + fold scales) ----
    for (int i = tid; i < 64 * 64; i += BLOCK) {
        int u = i >> 6, w = i & 63;
        wgtACT[w * 68 + u] = kSS0 * Wss0[i];
    }
    for (int i = tid; i < 64 * 32; i += BLOCK) {
        int u = i >> 5, w = i & 31;
        wgtACT[(64 + w) * 68 + u] = kSV1 * Wsv1[i];
    }
    for (int i = tid; i < 32 * 64; i += BLOCK) {
        int u = i >> 6, w = i & 63;
        wgtBT[w * 36 + u] = kVV0 * Wvv0[i];
    }
    for (int i = tid; i < 32 * 32; i += BLOCK) {
        int u = i >> 5, w = i & 31;
        wgtDT[w * 36 + u] = kVS1 * Wvs1[i];
    }

    // ---- Phase 1: per-wave activation staging ----
    const long long e0  = (long long)blockIdx.x * ROWS_PER_BLOCK + wave * 16;
    long long row = e0 + lr;
    if (row >= nrows) row = nrows - 1;  // clamp loads; stores are guarded later

    const float* xr = x1 + row * 160;
    float4 xx = *(const float4*)(x2 + row * 4);  // (s2, v2x, v2y, v2z)
    if (h == 0) *(float4*)&v2t[lr * 4] = xx;

    {   // s1 copy: half h handles u in [h*32, h*32+32)
        const int u0 = h * 32;
        #pragma unroll
        for (int u = 0; u < 32; u += 4) {
            float4 s = *(const float4*)(xr + u0 + u);
            *(float4*)&actS1[lr * 68 + u0 + u] = s;
        }
    }
    {   // v1 components + v1.v2 : half h handles u in [h*16, h*16+16)
        const int u0 = h * 16;
        #pragma unroll
        for (int u = 0; u < 16; ++u) {
            const int uu = u0 + u;
            float a = xr[64 + uu * 3 + 0];
            float b = xr[64 + uu * 3 + 1];
            float c = xr[64 + uu * 3 + 2];
            actV[0 * 576 + lr * 36 + uu] = a;
            actV[1 * 576 + lr * 36 + uu] = b;
            actV[2 * 576 + lr * 36 + uu] = c;
            actD[lr * 36 + uu] = a * xx.y + b * xx.z + c * xx.w;
        }
    }
    __syncthreads();

    const v8f vzero = {};
    const int h2 = 2 * h;

    // ---- GEMM A: S1 (16x64) @ [W_ss0 | W_sv1] (64x96) -> accA[0..3]=out0a, accA[4..5]=G ----
    v8f accA[6];
    {   // peeled k0 = 0 with inline C = 0
        v2f a = *(const v2f*)&actS1[lr * 68 + h2];
        #pragma unroll
        for (int n = 0; n < 6; ++n) {
            v2f b = *(const v2f*)&wgtACT[(n * 16 + lr) * 68 + h2];
            accA[n] = __builtin_amdgcn_wmma_f32_16x16x4_f32(
                false, a, false, b, (short)0, vzero, false, false);
        }
    }
    #pragma unroll 2
    for (int k0 = 4; k0 < 64; k0 += 4) {
        v2f a = *(const v2f*)&actS1[lr * 68 + k0 + h2];
        #pragma unroll
        for (int n = 0; n < 6; ++n) {
            v2f b = *(const v2f*)&wgtACT[(n * 16 + lr) * 68 + k0 + h2];
            accA[n] = __builtin_amdgcn_wmma_f32_16x16x4_f32(
                false, a, false, b, (short)0, accA[n], false, false);
        }
    }

    // ---- GEMM B: D (16x32) @ W_vv0 (32x64) -> accB[0..3] ----
    v8f accB[4];
    {   // peeled k0 = 0
        v2f a = *(const v2f*)&actD[lr * 36 + h2];
        #pragma unroll
        for (int n = 0; n < 4; ++n) {
            v2f b = *(const v2f*)&wgtBT[(n * 16 + lr) * 36 + h2];
            accB[n] = __builtin_amdgcn_wmma_f32_16x16x4_f32(
                false, a, false, b, (short)0, vzero, false, false);
        }
    }
    #pragma unroll 2
    for (int k0 = 4; k0 < 32; k0 += 4) {
        v2f a = *(const v2f*)&actD[lr * 36 + k0 + h2];
        #pragma unroll
        for (int n = 0; n < 4; ++n) {
            v2f b = *(const v2f*)&wgtBT[(n * 16 + lr) * 36 + k0 + h2];
            accB[n] = __builtin_amdgcn_wmma_f32_16x16x4_f32(
                false, a, false, b, (short)0, accB[n], false, false);
        }
    }

    // ---- Epilogue out0: out[b][w] = s2[b]*accA + accB + bias ----
    #pragma unroll
    for (int n = 0; n < 4; ++n) {
        float bw = bias0[n * 16 + lr];
        #pragma unroll
        for (int v = 0; v < 8; ++v) {
            int br = v + 8 * h;                 // batch row in tile (C/D layout)
            long long gr = e0 + br;
            if (gr < nrows) {
                float s2b = v2t[br * 4 + 0];
                out[gr * 160 + n * 16 + lr] = s2b * accA[n][v] + accB[n][v] + bw;
            }
        }
    }

    // ---- GEMM C (x3): V1_k (16x32) @ W_vs1 (32x32); B-fragment shared across k ----
    v8f accC0[2], accC1[2], accC2[2];
    {   // peeled k0 = 0
        v2f a0 = *(const v2f*)&actV[0 * 576 + lr * 36 + h2];
        v2f a1 = *(const v2f*)&actV[1 * 576 + lr * 36 + h2];
        v2f a2 = *(const v2f*)&actV[2 * 576 + lr * 36 + h2];
        #pragma unroll
        for (int n = 0; n < 2; ++n) {
            v2f b = *(const v2f*)&wgtDT[(n * 16 + lr) * 36 + h2];
            accC0[n] = __builtin_amdgcn_wmma_f32_16x16x4_f32(
                false, a0, false, b, (short)0, vzero, false, false);
            accC1[n] = __builtin_amdgcn_wmma_f32_16x16x4_f32(
                false, a1, false, b, (short)0, vzero, false, false);
            accC2[n] = __builtin_amdgcn_wmma_f32_16x16x4_f32(
                false, a2, false, b, (short)0, vzero, false, false);
        }
    }
    #pragma unroll 2
    for (int k0 = 4; k0 < 32; k0 += 4) {
        v2f a0 = *(const v2f*)&actV[0 * 576 + lr * 36 + k0 + h2];
        v2f a1 = *(const v2f*)&actV[1 * 576 + lr * 36 + k0 + h2];
        v2f a2 = *(const v2f*)&actV[2 * 576 + lr * 36 + k0 + h2];
        #pragma unroll
        for (int n = 0; n < 2; ++n) {
            v2f b = *(const v2f*)&wgtDT[(n * 16 + lr) * 36 + k0 + h2];
            accC0[n] = __builtin_amdgcn_wmma_f32_16x16x4_f32(
                false, a0, false, b, (short)0, accC0[n], false, false);
            accC1[n] = __builtin_amdgcn_wmma_f32_16x16x4_f32(
                false, a1, false, b, (short)0, accC1[n], false, false);
            accC2[n] = __builtin_amdgcn_wmma_f32_16x16x4_f32(
                false, a2, false, b, (short)0, accC2[n], false, false);
        }
    }

    // ---- Epilogue out1: out[b][64 + w*3 + k] = G[b][w]*v2[b][k] + s2[b]*F_k[b][w] ----
    #pragma unroll
    for (int v = 0; v < 8; ++v) {
        int br = v + 8 * h;
        long long gr = e0 + br;
        if (gr < nrows) {
            float s2b = v2t[br * 4 + 0];
            float vx  = v2t[br * 4 + 1];
            float vy  = v2t[br * 4 + 2];
            float vz  = v2t[br * 4 + 3];
            #pragma unroll
            for (int n = 0; n < 2; ++n) {
                float g = accA[4 + n][v];
                float* ob = out + gr * 160 + 64 + (long long)(n * 16 + lr) * 3;
                ob[0] = g * vx + s2b * accC0[n][v];
                ob[1] = g * vy + s2b * accC1[n][v];
                ob[2] = g * vz + s2b * accC2[n][v];
            }
        }
    }
}

extern "C" void kernel_launch(void* const* d_in, const int* in_sizes, int n_in,
                              void* d_out, int out_size, void* d_ws, size_t ws_size,
                              hipStream_t stream) {
    const float* x1    = (const float*)d_in[0];
    const float* x2    = (const float*)d_in[1];
    const float* Wss0  = (const float*)d_in[2];
    const float* Wvv0  = (const float*)d_in[3];
    const float* Wsv1  = (const float*)d_in[4];
    const float* Wvs1  = (const float*)d_in[5];
    const float* bias0 = (const float*)d_in[6];
    float* out = (float*)d_out;

    const int nrows  = in_sizes[0] / 160;                              // N (=400000)
    const int blocks = (nrows + ROWS_PER_BLOCK - 1) / ROWS_PER_BLOCK;  // 6250

    o3tp_wmma_f32_kernel<<<blocks, BLOCK, 0, stream>>>(
        x1, x2, Wss0, Wvv0, Wsv1, Wvs1, bias0, out, nrows);
}